// HGNN_85993835200538
// MI455X (gfx1250) — compile-verified
//
#include <hip/hip_runtime.h>
#include <type_traits>

// ---------------------------------------------------------------------------
// HGNN pipeline for MI455X (gfx1250, wave32).
// Dense GEMMs via V_WMMA_F32_16X16X4_F32 with MT-way M register blocking;
// sparse Laplacian applies via wave-per-nnz vectorized scatters with native
// fp32 global atomics. Memory-bound: scaling fused into scatters, concat
// fused into GEMM A-tile staging, all bulk moves in b128.
// ---------------------------------------------------------------------------

typedef float v2f __attribute__((ext_vector_type(2)));
typedef float v4f __attribute__((ext_vector_type(4)));
typedef float v8f __attribute__((ext_vector_type(8)));

static constexpr int cN0 = 100000, cN1 = 25000, cN2 = 6250;
static constexpr int cE0 = 20000,  cE1 = 5000,  cE2 = 1250;
static constexpr int cNNZ0 = 1600000, cNNZ1 = 400000, cNNZ2 = 100000;

__device__ __forceinline__ void atomAddF(float* p, float v) {
  // lowers to global_atomic_add_f32 (relaxed, device scope)
  __hip_atomic_fetch_add(p, v, __ATOMIC_RELAXED, __HIP_MEMORY_SCOPE_AGENT);
}

// ------------------------------- utility -----------------------------------

__global__ void k_zero(float* __restrict__ p, long n) {
  long t = (long)blockIdx.x * blockDim.x + threadIdx.x;
  long stride = (long)gridDim.x * blockDim.x;
  for (; t < n; t += stride) p[t] = 0.0f;
}

__global__ void k_zero4(float* __restrict__ p, long n4) {  // n4 = count of float4
  long t = (long)blockIdx.x * blockDim.x + threadIdx.x;
  long stride = (long)gridDim.x * blockDim.x;
  v4f z = 0.0f;
  for (; t < n4; t += stride) *(v4f*)(p + t * 4) = z;
}

__global__ void k_count_deg(const int* __restrict__ vi, const int* __restrict__ ei,
                            float* __restrict__ dV, float* __restrict__ dE, int nnz) {
  int t = blockIdx.x * blockDim.x + threadIdx.x;
  int stride = gridDim.x * blockDim.x;
  for (; t < nnz; t += stride) {
    atomAddF(&dV[vi[t]], 1.0f);
    atomAddF(&dE[ei[t]], 1.0f);
  }
}

__global__ void k_count_assign(const int* __restrict__ assign, float* __restrict__ cnt, int n) {
  int t = blockIdx.x * blockDim.x + threadIdx.x;
  int stride = gridDim.x * blockDim.x;
  for (; t < n; t += stride) atomAddF(&cnt[assign[t]], 1.0f);
}

__global__ void k_finalize_rsqrt(float* __restrict__ d, int n) {
  int t = blockIdx.x * blockDim.x + threadIdx.x;
  if (t < n) d[t] = (d[t] > 0.0f) ? rsqrtf(d[t]) : 0.0f;
}

__global__ void k_finalize_recip(float* __restrict__ d, int n) {
  int t = blockIdx.x * blockDim.x + threadIdx.x;
  if (t < n) d[t] = (d[t] > 0.0f) ? (1.0f / d[t]) : 0.0f;
}

// ----------------------- sparse Laplacian pieces ---------------------------
// One wave per nnz entry: indices loaded once per wave, feature row moved as
// b128 (D=128) / b64 (D=64) vector loads, scalar fp32 atomics for the scatter.

// Y[e] += X[v] * dv_is[v]
template <int D>
__global__ void k_scatter_v2e(const float* __restrict__ X, const float* __restrict__ dvis,
                              const int* __restrict__ vi, const int* __restrict__ ei,
                              float* __restrict__ Y, int nnz) {
  constexpr int VPL = D / 32;
  using VT = std::conditional_t<VPL == 4, v4f, v2f>;
  long gt = (long)blockIdx.x * blockDim.x + threadIdx.x;
  int j = (int)(gt >> 5);
  int lane = threadIdx.x & 31;
  if (j >= nnz) return;
  int v = vi[j], e = ei[j];
  float s = dvis[v];
  VT t = *(const VT*)(X + (size_t)v * D + lane * VPL);
  float* yr = Y + (size_t)e * D + lane * VPL;
#pragma unroll
  for (int u = 0; u < VPL; ++u) atomAddF(yr + u, t[u] * s);
}

// Z[v] += Y[e] * de_i[e]
template <int D>
__global__ void k_scatter_e2v(const float* __restrict__ Y, const float* __restrict__ dei,
                              const int* __restrict__ vi, const int* __restrict__ ei,
                              float* __restrict__ Z, int nnz) {
  constexpr int VPL = D / 32;
  using VT = std::conditional_t<VPL == 4, v4f, v2f>;
  long gt = (long)blockIdx.x * blockDim.x + threadIdx.x;
  int j = (int)(gt >> 5);
  int lane = threadIdx.x & 31;
  if (j >= nnz) return;
  int v = vi[j], e = ei[j];
  float s = dei[e];
  VT t = *(const VT*)(Y + (size_t)e * D + lane * VPL);
  float* zr = Z + (size_t)v * D + lane * VPL;
#pragma unroll
  for (int u = 0; u < VPL; ++u) atomAddF(zr + u, t[u] * s);
}

// Z[i] *= s[i] (+ optional ReLU); also pool mean with s = 1/cnt. float4 path.
template <int D>
__global__ void k_scale_rows(float* __restrict__ Z, const float* __restrict__ s, int n, int relu) {
  long t4 = (long)blockIdx.x * blockDim.x + threadIdx.x;
  long total4 = (long)n * (D / 4);
  if (t4 >= total4) return;
  int i = (int)(t4 / (D / 4));
  v4f val = *(v4f*)(Z + t4 * 4) * s[i];
  if (relu) {
#pragma unroll
    for (int u = 0; u < 4; ++u) val[u] = fmaxf(val[u], 0.0f);
  }
  *(v4f*)(Z + t4 * 4) = val;
}

// Xc[assign[j]] += X[j]  (wave per fine row)
template <int D>
__global__ void k_scatter_pool(const float* __restrict__ X, const int* __restrict__ assign,
                               float* __restrict__ Xc, int n) {
  constexpr int VPL = D / 32;
  using VT = std::conditional_t<VPL == 4, v4f, v2f>;
  long gt = (long)blockIdx.x * blockDim.x + threadIdx.x;
  int j = (int)(gt >> 5);
  int lane = threadIdx.x & 31;
  if (j >= n) return;
  int c = assign[j];
  VT t = *(const VT*)(X + (size_t)j * D + lane * VPL);
  float* xr = Xc + (size_t)c * D + lane * VPL;
#pragma unroll
  for (int u = 0; u < VPL; ++u) atomAddF(xr + u, t[u]);
}

// ------------------------------ WMMA GEMM ----------------------------------
// C[M x N] = A[M x K] @ W[K x N] + bias.
// blockDim.x = 32*(N/16) waves, each wave owns one 16-col N tile and MT
// 16-row M tiles (register blocking -> each B fetch feeds MT WMMAs, and the
// MT independent accumulators hide the WMMA->WMMA RAW hazard).
// Bounds checks are uniform per block: all blocks but the last take
// straight-line (no exec-mask) staging/store paths.
//
// f32 16x16x4 layouts (ISA 7.12.2):
//   A: lane l, vgpr v -> A[l&15][k0 + 2*(l>>4) + v]
//   B: lane l, vgpr v -> W[k0 + 2*(l>>4) + v][ntile*16 + (l&15)]
//   D: lane l, vgpr r -> C[row0 + r + 8*(l>>4)][ntile*16 + (l&15)]
template <int K, int N, int MT>
__global__ void k_gemm_wmma(const float* __restrict__ A, const float* __restrict__ W,
                            const float* __restrict__ bias, float* __restrict__ C, int M) {
  constexpr int LK = K + 4;  // pad: stride (K+4)%64 = 4 banks -> conflict-free
  __shared__ float lds[16 * MT * LK];
  const int lane = threadIdx.x & 31;
  const int wave = threadIdx.x >> 5;  // n-tile index
  const int lo = lane & 15;
  const int hi = lane >> 4;
  const int row0 = blockIdx.x * (16 * MT);
  const bool fullTile = (row0 + 16 * MT) <= M;  // uniform (scalar) predicate

  // stage A (16*MT x K) into LDS with b128 loads/stores
  if (fullTile) {
    for (int idx = threadIdx.x; idx < 16 * MT * (K / 4); idx += blockDim.x) {
      int r = idx / (K / 4), cq = idx % (K / 4);
      *(v4f*)&lds[r * LK + cq * 4] = *(const v4f*)(A + (size_t)(row0 + r) * K + cq * 4);
    }
  } else {
    for (int idx = threadIdx.x; idx < 16 * MT * (K / 4); idx += blockDim.x) {
      int r = idx / (K / 4), cq = idx % (K / 4);
      int m = row0 + r;
      if (m >= M) m = M - 1;  // clamp loads; stores are guarded
      *(v4f*)&lds[r * LK + cq * 4] = *(const v4f*)(A + (size_t)m * K + cq * 4);
    }
  }
  __syncthreads();

  const int n = wave * 16 + lo;
  float bv = bias[n];
  v8f acc[MT];
#pragma unroll
  for (int mt = 0; mt < MT; ++mt)
#pragma unroll
    for (int r = 0; r < 8; ++r) acc[mt][r] = bv;

  for (int k0 = 0; k0 < K; k0 += 4) {
    v2f b;
#pragma unroll
    for (int u = 0; u < 2; ++u) b[u] = W[(size_t)(k0 + 2 * hi + u) * N + n];
#pragma unroll
    for (int mt = 0; mt < MT; ++mt) {
      v2f a;
#pragma unroll
      for (int u = 0; u < 2; ++u) a[u] = lds[(mt * 16 + lo) * LK + k0 + 2 * hi + u];
      acc[mt] = __builtin_amdgcn_wmma_f32_16x16x4_f32(false, a, false, b, (short)0,
                                                      acc[mt], false, false);
    }
  }

  if (fullTile) {
#pragma unroll
    for (int mt = 0; mt < MT; ++mt)
#pragma unroll
      for (int r = 0; r < 8; ++r)
        C[(size_t)(row0 + mt * 16 + r + 8 * hi) * N + n] = acc[mt][r];
  } else {
#pragma unroll
    for (int mt = 0; mt < MT; ++mt)
#pragma unroll
      for (int r = 0; r < 8; ++r) {
        int m = row0 + mt * 16 + r + 8 * hi;
        if (m < M) C[(size_t)m * N + n] = acc[mt][r];
      }
  }
}

// C[M x N] = concat(Acoarse[assign[m]], Afine[m]) @ W[256 x N] + bias.
// Gather + concat fused into A-tile staging (no Xu materialization).
template <int N, int MT>
__global__ void k_gemm_concat_wmma(const float* __restrict__ Acoarse,
                                   const int* __restrict__ assign,
                                   const float* __restrict__ Afine,
                                   const float* __restrict__ W, const float* __restrict__ bias,
                                   float* __restrict__ C, int M) {
  constexpr int K = 256;
  constexpr int LK = K + 4;
  __shared__ float lds[16 * MT * LK];
  const int lane = threadIdx.x & 31;
  const int wave = threadIdx.x >> 5;
  const int lo = lane & 15;
  const int hi = lane >> 4;
  const int row0 = blockIdx.x * (16 * MT);
  const bool fullTile = (row0 + 16 * MT) <= M;

  for (int idx = threadIdx.x; idx < 16 * MT * 64; idx += blockDim.x) {
    int r = idx >> 6, cq = idx & 63;  // 64 float4 per 256-wide row
    int m = row0 + r;
    if (!fullTile && m >= M) m = M - 1;
    v4f t = (cq < 32) ? *(const v4f*)(Acoarse + (size_t)assign[m] * 128 + cq * 4)
                      : *(const v4f*)(Afine + (size_t)m * 128 + (cq - 32) * 4);
    *(v4f*)&lds[r * LK + cq * 4] = t;
  }
  __syncthreads();

  const int n = wave * 16 + lo;
  float bv = bias[n];
  v8f acc[MT];
#pragma unroll
  for (int mt = 0; mt < MT; ++mt)
#pragma unroll
    for (int r = 0; r < 8; ++r) acc[mt][r] = bv;

  for (int k0 = 0; k0 < K; k0 += 4) {
    v2f b;
#pragma unroll
    for (int u = 0; u < 2; ++u) b[u] = W[(size_t)(k0 + 2 * hi + u) * N + n];
#pragma unroll
    for (int mt = 0; mt < MT; ++mt) {
      v2f a;
#pragma unroll
      for (int u = 0; u < 2; ++u) a[u] = lds[(mt * 16 + lo) * LK + k0 + 2 * hi + u];
      acc[mt] = __builtin_amdgcn_wmma_f32_16x16x4_f32(false, a, false, b, (short)0,
                                                      acc[mt], false, false);
    }
  }

  if (fullTile) {
#pragma unroll
    for (int mt = 0; mt < MT; ++mt)
#pragma unroll
      for (int r = 0; r < 8; ++r)
        C[(size_t)(row0 + mt * 16 + r + 8 * hi) * N + n] = acc[mt][r];
  } else {
#pragma unroll
    for (int mt = 0; mt < MT; ++mt)
#pragma unroll
      for (int r = 0; r < 8; ++r) {
        int m = row0 + mt * 16 + r + 8 * hi;
        if (m < M) C[(size_t)m * N + n] = acc[mt][r];
      }
  }
}

// ------------------------------- host side ---------------------------------

static inline int cdiv(long a, long b) { return (int)((a + b - 1) / b); }

template <int D>
static void l_apply(const float* Xin, float* Zout, float* Ytmp,
                    const float* dvis, const float* dei,
                    const int* vi, const int* ei,
                    int n, int e, int nnz, int relu, hipStream_t s) {
  k_zero4<<<cdiv((long)e * D / 4, 256), 256, 0, s>>>(Ytmp, (long)e * D / 4);
  k_zero4<<<cdiv((long)n * D / 4, 256), 256, 0, s>>>(Zout, (long)n * D / 4);
  long waveThreads = (long)nnz * 32;
  k_scatter_v2e<D><<<cdiv(waveThreads, 256), 256, 0, s>>>(Xin, dvis, vi, ei, Ytmp, nnz);
  k_scatter_e2v<D><<<cdiv(waveThreads, 256), 256, 0, s>>>(Ytmp, dei, vi, ei, Zout, nnz);
  k_scale_rows<D><<<cdiv((long)n * D / 4, 256), 256, 0, s>>>(Zout, dvis, n, relu);
}

extern "C" void kernel_launch(void* const* d_in, const int* in_sizes, int n_in,
                              void* d_out, int out_size, void* d_ws, size_t ws_size,
                              hipStream_t stream) {
  (void)in_sizes; (void)n_in; (void)out_size; (void)ws_size;
  const float* X  = (const float*)d_in[0];
  const float* W0 = (const float*)d_in[1];
  const float* b0 = (const float*)d_in[2];
  const float* W1 = (const float*)d_in[3];
  const float* b1 = (const float*)d_in[4];
  const float* W2 = (const float*)d_in[5];
  const float* b2 = (const float*)d_in[6];
  const float* W3 = (const float*)d_in[7];
  const float* b3 = (const float*)d_in[8];
  const float* W4 = (const float*)d_in[9];
  const float* b4 = (const float*)d_in[10];
  const int* H0v = (const int*)d_in[11];
  const int* H0e = (const int*)d_in[12];
  const int* H1v = (const int*)d_in[13];
  const int* H1e = (const int*)d_in[14];
  const int* H2v = (const int*)d_in[15];
  const int* H2e = (const int*)d_in[16];
  const int* as0 = (const int*)d_in[17];
  const int* as1 = (const int*)d_in[18];
  float* out = (float*)d_out;

  // workspace bump allocator (~143 MB)
  float* w = (float*)d_ws;
  size_t o = 0;
  auto alloc = [&](size_t n) { float* p = w + o; o += n; return p; };
  float* dv0  = alloc(cN0);   // Dv^-1/2 level 0
  float* de0  = alloc(cE0);   // De^-1   level 0
  float* dv1  = alloc(cN1);
  float* de1  = alloc(cE1);
  float* dv2  = alloc(cN2);
  float* de2  = alloc(cE2);
  float* cnt0 = alloc(cN1);   // becomes 1/cluster-size (level 0 -> 1)
  float* cnt1 = alloc(cN2);
  float* h0   = alloc((size_t)cN0 * 128);
  float* h1   = alloc((size_t)cN1 * 128);
  float* h2   = alloc((size_t)cN2 * 128);
  float* tmpE = alloc((size_t)cE0 * 128);  // edge temp, max level-0 size
  float* g    = alloc((size_t)cN0 * 128);  // GEMM output scratch
  float* pool = alloc((size_t)cN1 * 128);  // pooled features / u1 scratch

  // ---- degrees / pooling counts (zero the contiguous region once) ----
  long degLen = (long)cN0 + cE0 + cN1 + cE1 + cN2 + cE2 + cN1 + cN2;
  k_zero<<<cdiv(degLen, 256), 256, 0, stream>>>(dv0, degLen);
  k_count_deg<<<cdiv(cNNZ0, 256), 256, 0, stream>>>(H0v, H0e, dv0, de0, cNNZ0);
  k_count_deg<<<cdiv(cNNZ1, 256), 256, 0, stream>>>(H1v, H1e, dv1, de1, cNNZ1);
  k_count_deg<<<cdiv(cNNZ2, 256), 256, 0, stream>>>(H2v, H2e, dv2, de2, cNNZ2);
  k_count_assign<<<cdiv(cN0, 256), 256, 0, stream>>>(as0, cnt0, cN0);
  k_count_assign<<<cdiv(cN1, 256), 256, 0, stream>>>(as1, cnt1, cN1);
  k_finalize_rsqrt<<<cdiv(cN0, 256), 256, 0, stream>>>(dv0, cN0);
  k_finalize_rsqrt<<<cdiv(cN1, 256), 256, 0, stream>>>(dv1, cN1);
  k_finalize_rsqrt<<<cdiv(cN2, 256), 256, 0, stream>>>(dv2, cN2);
  k_finalize_recip<<<cdiv(cE0, 256), 256, 0, stream>>>(de0, cE0);
  k_finalize_recip<<<cdiv(cE1, 256), 256, 0, stream>>>(de1, cE1);
  k_finalize_recip<<<cdiv(cE2, 256), 256, 0, stream>>>(de2, cE2);
  k_finalize_recip<<<cdiv(cN1, 256), 256, 0, stream>>>(cnt0, cN1);
  k_finalize_recip<<<cdiv(cN2, 256), 256, 0, stream>>>(cnt1, cN2);

  // ---- level 0: h0 = relu(L0(X@W0 + b0)) ----
  k_gemm_wmma<128, 128, 4><<<cdiv(cN0, 64), 256, 0, stream>>>(X, W0, b0, g, cN0);
  l_apply<128>(g, h0, tmpE, dv0, de0, H0v, H0e, cN0, cE0, cNNZ0, 1, stream);

  // ---- pool 0 -> 1 ----
  k_zero4<<<cdiv((long)cN1 * 32, 256), 256, 0, stream>>>(pool, (long)cN1 * 32);
  k_scatter_pool<128><<<cdiv((long)cN0 * 32, 256), 256, 0, stream>>>(h0, as0, pool, cN0);
  k_scale_rows<128><<<cdiv((long)cN1 * 32, 256), 256, 0, stream>>>(pool, cnt0, cN1, 0);

  // ---- level 1: h1 = relu(L1(Xc@W1 + b1)) ----
  k_gemm_wmma<128, 128, 4><<<cdiv(cN1, 64), 256, 0, stream>>>(pool, W1, b1, g, cN1);
  l_apply<128>(g, h1, tmpE, dv1, de1, H1v, H1e, cN1, cE1, cNNZ1, 1, stream);

  // ---- pool 1 -> 2 ----
  k_zero4<<<cdiv((long)cN2 * 32, 256), 256, 0, stream>>>(pool, (long)cN2 * 32);
  k_scatter_pool<128><<<cdiv((long)cN1 * 32, 256), 256, 0, stream>>>(h1, as1, pool, cN1);
  k_scale_rows<128><<<cdiv((long)cN2 * 32, 256), 256, 0, stream>>>(pool, cnt1, cN2, 0);

  // ---- coarsest: h2 = relu(L2(Xc@W2 + b2)) ----
  k_gemm_wmma<128, 128, 4><<<cdiv(cN2, 64), 256, 0, stream>>>(pool, W2, b2, g, cN2);
  l_apply<128>(g, h2, tmpE, dv2, de2, H2v, H2e, cN2, cE2, cNNZ2, 1, stream);

  // ---- up 1: u1 = relu(L1(concat(h2[assign1], h1) @ W3 + b3)) ----
  k_gemm_concat_wmma<128, 2><<<cdiv(cN1, 32), 256, 0, stream>>>(h2, as1, h1, W3, b3, g, cN1);
  l_apply<128>(g, pool, tmpE, dv1, de1, H1v, H1e, cN1, cE1, cNNZ1, 1, stream);  // u1 in pool

  // ---- up 0: out = L0(concat(u1[assign0], h0) @ W4 + b4) ----
  k_gemm_concat_wmma<64, 2><<<cdiv(cN0, 32), 128, 0, stream>>>(pool, as0, h0, W4, b4, g, cN0);
  l_apply<64>(g, out, tmpE, dv0, de0, H0v, H0e, cN0, cE0, cNNZ0, 0, stream);
}